// GTConv_42614665511411
// MI455X (gfx1250) — compile-verified
//
#include <hip/hip_runtime.h>
#include <hip/hip_bf16.h>
#include <stdint.h>

// ---------------------------------------------------------------------------
// GTConv for MI455X (gfx1250, wave32).
// Sort-dominated memory-bound pipeline; channel mixing (softmax(W) @ S) done
// with v_wmma_f32_16x16x4_f32 (CDNA5 WMMA).
// ---------------------------------------------------------------------------

typedef float v2f __attribute__((ext_vector_type(2)));
typedef float v8f __attribute__((ext_vector_type(8)));

static constexpr long long C_IN    = 5;
static constexpr long long C_OUT   = 4;
static constexpr long long E_EDGES = 2000000;
static constexpr long long N_NODES = 100000;
static constexpr long long K_TOT   = C_IN * E_EDGES;       // 10,000,000 (multiple of 16)

static constexpr int RADIX_BITS = 9;
static constexpr int RADIX      = 1 << RADIX_BITS;         // 512
static constexpr int N_PASSES   = 4;                       // 4*9 = 36 >= 34 key bits
static constexpr int RS_THREADS = 256;                     // 8 wave32
static constexpr int RS_VT      = 8;
static constexpr int RS_CHUNK   = RS_THREADS * RS_VT;      // 2048
static constexpr int RS_BLOCKS  = (int)((K_TOT + RS_CHUNK - 1) / RS_CHUNK); // 4883
static constexpr long long HIST_N = (long long)RADIX * RS_BLOCKS;           // 2,500,096

static constexpr int SC_THREADS = 256;
static constexpr int SC_VT      = 8;
static constexpr int SC_CHUNK   = SC_THREADS * SC_VT;      // 2048

// Workspace layout (bytes). S aliases the radix "B" ping buffer, which is dead
// after the final (4th) pass writes back into buffer A. Total ~330 MB.
static constexpr size_t OFF_KEYS0  = 0;                                   // K*8 = 80 MB
static constexpr size_t OFF_IDX0   = OFF_KEYS0 + (size_t)K_TOT * 8;       // K*4 = 40 MB
static constexpr size_t OFF_KEYS1  = OFF_IDX0  + (size_t)K_TOT * 4;       // 80 MB
static constexpr size_t OFF_IDX1   = OFF_KEYS1 + (size_t)K_TOT * 8;       // 40 MB
static constexpr size_t OFF_S      = OFF_KEYS1;                           // 5*K*4 = 200 MB (alias)
static constexpr size_t OFF_HIST   = OFF_S + (size_t)C_IN * K_TOT * 4;    // 320,000,000
static constexpr size_t OFF_AUX    = OFF_HIST   + (size_t)HIST_N * 4;     // scan partials (<=1221)
static constexpr size_t OFF_SEGCNT = OFF_AUX    + 8192;
static constexpr size_t OFF_SEGAUX = OFF_SEGCNT + 20480;
static constexpr size_t OFF_WWS    = OFF_SEGAUX + 256;

#ifdef __has_builtin
#if __has_builtin(__builtin_amdgcn_ballot_w32)
#define BALLOT32(p) __builtin_amdgcn_ballot_w32((bool)(p))
#endif
#endif
#ifndef BALLOT32
#define BALLOT32(p) ((unsigned)__ballot((int)(p)))
#endif

// ---------------------------------------------------------------------------
// 1) softmax over weight rows (4x5); writes to ws (for GEMM) and to d_out tail
// ---------------------------------------------------------------------------
__global__ void gt_softmax_kernel(const float* __restrict__ weight,
                                  float* __restrict__ w_ws,
                                  float* __restrict__ w_out) {
  int o = threadIdx.x;
  if (o < (int)C_OUT) {
    float m = -1e30f;
    for (int c = 0; c < (int)C_IN; ++c) m = fmaxf(m, weight[o * C_IN + c]);
    float ex[(int)C_IN]; float s = 0.f;
    for (int c = 0; c < (int)C_IN; ++c) { ex[c] = expf(weight[o * C_IN + c] - m); s += ex[c]; }
    float inv = 1.f / s;
    for (int c = 0; c < (int)C_IN; ++c) {
      float v = ex[c] * inv;
      w_ws[o * C_IN + c]  = v;
      w_out[o * C_IN + c] = v;
    }
  }
}

// ---------------------------------------------------------------------------
// 2) key generation: key[k] = src*N + dst (34-bit), idx[k] = k
// ---------------------------------------------------------------------------
__global__ void gt_keygen_kernel(const int* __restrict__ edge_index,
                                 unsigned long long* __restrict__ keys,
                                 unsigned* __restrict__ idx) {
  long long j = (long long)blockIdx.x * blockDim.x + threadIdx.x;
  if (j >= K_TOT) return;
  long long c = j / E_EDGES;
  long long e = j - c * E_EDGES;
  unsigned src = (unsigned)edge_index[c * 2 * E_EDGES + e];
  unsigned dst = (unsigned)edge_index[c * 2 * E_EDGES + E_EDGES + e];
  keys[j] = (unsigned long long)src * (unsigned long long)N_NODES + dst;
  idx[j]  = (unsigned)j;
}

// ---------------------------------------------------------------------------
// 3a) per-block digit histograms, digit-major layout hist[d * RS_BLOCKS + b]
// ---------------------------------------------------------------------------
__global__ void gt_radix_hist_kernel(const unsigned long long* __restrict__ srcK,
                                     unsigned* __restrict__ hist, int shift) {
  __shared__ unsigned h[RADIX];
  int tid = threadIdx.x;
  for (int i = tid; i < RADIX; i += RS_THREADS) h[i] = 0u;
  __syncthreads();
  long long base = (long long)blockIdx.x * RS_CHUNK;
  __builtin_prefetch(srcK + base + RS_CHUNK, 0, 0);  // global_prefetch_b8
  for (int v = 0; v < RS_VT; ++v) {
    long long j = base + (long long)v * RS_THREADS + tid;
    if (j < K_TOT) {
      unsigned d = (unsigned)((srcK[j] >> shift) & (RADIX - 1));
      atomicAdd(&h[d], 1u);
    }
  }
  __syncthreads();
  for (int d = tid; d < RADIX; d += RS_THREADS)
    hist[(size_t)d * RS_BLOCKS + blockIdx.x] = h[d];
}

// ---------------------------------------------------------------------------
// generic two-level exclusive scan (uint32): block scan -> aux scan -> add
// ---------------------------------------------------------------------------
__global__ void gt_scan_block_kernel(const unsigned* __restrict__ in,
                                     unsigned* __restrict__ out,
                                     unsigned* __restrict__ aux, long long n) {
  __shared__ unsigned tsum[SC_THREADS];
  int tid = threadIdx.x;
  long long i0 = (long long)blockIdx.x * SC_CHUNK + (long long)tid * SC_VT;
  unsigned v[SC_VT];
#pragma unroll
  for (int j = 0; j < SC_VT; ++j) { long long i = i0 + j; v[j] = (i < n) ? in[i] : 0u; }
  unsigned run = 0;
#pragma unroll
  for (int j = 0; j < SC_VT; ++j) { unsigned t = v[j]; v[j] = run; run += t; }
  tsum[tid] = run;
  __syncthreads();
  for (int off = 1; off < SC_THREADS; off <<= 1) {
    unsigned t = (tid >= off) ? tsum[tid - off] : 0u;
    __syncthreads();
    if (tid >= off) tsum[tid] += t;
    __syncthreads();
  }
  unsigned texcl = tid ? tsum[tid - 1] : 0u;
  if (tid == 0) aux[blockIdx.x] = tsum[SC_THREADS - 1];
#pragma unroll
  for (int j = 0; j < SC_VT; ++j) { long long i = i0 + j; if (i < n) out[i] = v[j] + texcl; }
}

__global__ void gt_scan_aux_kernel(unsigned* __restrict__ data, int n) {
  __shared__ unsigned buf[1024];
  int tid = threadIdx.x;
  unsigned carry = 0;
  for (int base = 0; base < n; base += 1024) {
    int i = base + tid;
    buf[tid] = (i < n) ? data[i] : 0u;
    __syncthreads();
    for (int off = 1; off < 1024; off <<= 1) {
      unsigned t = (tid >= off) ? buf[tid - off] : 0u;
      __syncthreads();
      if (tid >= off) buf[tid] += t;
      __syncthreads();
    }
    unsigned excl  = (tid ? buf[tid - 1] : 0u) + carry;
    unsigned total = buf[1023];
    __syncthreads();
    if (i < n) data[i] = excl;
    carry += total;
  }
}

__global__ void gt_scan_add_kernel(unsigned* __restrict__ data,
                                   const unsigned* __restrict__ aux, long long n) {
  long long base = (long long)blockIdx.x * SC_CHUNK;
  unsigned add = aux[blockIdx.x];
  for (int j = threadIdx.x; j < SC_CHUNK; j += SC_THREADS) {
    long long i = base + j;
    if (i < n) data[i] += add;
  }
}

// ---------------------------------------------------------------------------
// 3b) stable radix scatter: wave32 ballot peer-ranking + LDS wave counters
// ---------------------------------------------------------------------------
__global__ void gt_radix_scatter_kernel(const unsigned long long* __restrict__ srcK,
                                        const unsigned* __restrict__ srcI,
                                        unsigned long long* __restrict__ dstK,
                                        unsigned* __restrict__ dstI,
                                        const unsigned* __restrict__ histScan, int shift) {
  __shared__ unsigned waveCnt[8][RADIX];   // per-wave running digit counts
  __shared__ unsigned digitBase[RADIX];    // global base for this block
  int tid = threadIdx.x, lane = tid & 31, wv = tid >> 5;

  for (int d = tid; d < RADIX; d += RS_THREADS)
    digitBase[d] = histScan[(size_t)d * RS_BLOCKS + blockIdx.x];
  for (int i = tid; i < 8 * RADIX; i += RS_THREADS) (&waveCnt[0][0])[i] = 0u;
  __syncthreads();

  long long base = (long long)blockIdx.x * RS_CHUNK;
  unsigned long long keyv[RS_VT];
  unsigned idxv[RS_VT], rnk[RS_VT];
  unsigned okbits = 0;

  // item order within block == memory order: (wave, slot, lane)
#pragma unroll
  for (int v = 0; v < RS_VT; ++v) {
    long long j = base + (long long)wv * (32 * RS_VT) + (long long)v * 32 + lane;
    bool ok = j < K_TOT;
    keyv[v] = ok ? srcK[j] : 0ull;
    idxv[v] = ok ? srcI[j] : 0u;
    if (ok) okbits |= 1u << v;
  }

#pragma unroll
  for (int v = 0; v < RS_VT; ++v) {
    bool ok = (okbits >> v) & 1u;
    unsigned d = ok ? (unsigned)((keyv[v] >> shift) & (RADIX - 1)) : 0u;
    unsigned vm = BALLOT32(ok);
    unsigned peers = vm;
#pragma unroll
    for (int b = 0; b < RADIX_BITS; ++b) {
      unsigned bm = BALLOT32((d >> b) & 1u);
      peers &= ((d >> b) & 1u) ? bm : ~bm;
    }
    unsigned lt = (1u << lane) - 1u;
    unsigned r = __popc(peers & lt);
    unsigned total = __popc(peers);
    int leader = peers ? (__ffs((int)peers) - 1) : 0;
    unsigned old = 0;
    if (ok && lane == leader) old = atomicAdd(&waveCnt[wv][d], total);
    old = __shfl(old, leader);             // ds_bpermute broadcast from leader
    rnk[v] = old + r;
  }
  __syncthreads();

  // exclusive prefix across waves per digit
  for (int d = tid; d < RADIX; d += RS_THREADS) {
    unsigned run = 0;
    for (int w = 0; w < 8; ++w) { unsigned t = waveCnt[w][d]; waveCnt[w][d] = run; run += t; }
  }
  __syncthreads();

#pragma unroll
  for (int v = 0; v < RS_VT; ++v) {
    if ((okbits >> v) & 1u) {
      unsigned d = (unsigned)((keyv[v] >> shift) & (RADIX - 1));
      unsigned pos = digitBase[d] + waveCnt[wv][d] + rnk[v];
      dstK[pos] = keyv[v];
      dstI[pos] = idxv[v];
    }
  }
}

// ---------------------------------------------------------------------------
// 4a) per-block count of segment heads
// ---------------------------------------------------------------------------
__global__ void gt_segcount_kernel(const unsigned long long* __restrict__ keys,
                                   unsigned* __restrict__ blockCounts) {
  __shared__ unsigned red[RS_THREADS];
  int tid = threadIdx.x;
  long long base = (long long)blockIdx.x * RS_CHUNK;
  unsigned cnt = 0;
  for (int v = 0; v < RS_VT; ++v) {
    long long j = base + (long long)v * RS_THREADS + tid;
    if (j < K_TOT) {
      bool f = (j == 0) || (keys[j] != keys[j - 1]);
      cnt += f ? 1u : 0u;
    }
  }
  red[tid] = cnt;
  __syncthreads();
  for (int off = RS_THREADS / 2; off > 0; off >>= 1) {
    if (tid < off) red[tid] += red[tid + off];
    __syncthreads();
  }
  if (tid == 0) blockCounts[blockIdx.x] = red[0];
}

// ---------------------------------------------------------------------------
// 4b) fused: seg_id via ballot prefix; atomic scatter ev -> S; index output
// ---------------------------------------------------------------------------
__global__ void gt_final_scatter_kernel(const unsigned long long* __restrict__ keys,
                                        const unsigned* __restrict__ idx,
                                        const float* __restrict__ edge_value,
                                        const unsigned* __restrict__ blockBaseScan,
                                        float* __restrict__ S,
                                        int* __restrict__ outIndex) {
  __shared__ unsigned waveTot[8];
  __shared__ unsigned waveExcl[8];
  int tid = threadIdx.x, lane = tid & 31, wv = tid >> 5;
  long long base = (long long)blockIdx.x * RS_CHUNK;

  unsigned long long keyv[RS_VT];
  unsigned inc[RS_VT];
  unsigned okbits = 0, flagbits = 0, cnt = 0;

#pragma unroll
  for (int v = 0; v < RS_VT; ++v) {
    long long j = base + (long long)wv * (32 * RS_VT) + (long long)v * 32 + lane;
    bool ok = j < K_TOT;
    unsigned long long k = ok ? keys[j] : 0ull;
    bool f = ok && ((j == 0) || (k != keys[j - 1]));
    keyv[v] = k;
    if (ok) okbits |= 1u << v;
    if (f)  flagbits |= 1u << v;
    unsigned fb = BALLOT32(f);
    unsigned le = (2u << lane) - 1u;    // lane==31 -> 0xffffffff
    inc[v] = cnt + __popc(fb & le);     // inclusive count within wave
    cnt += __popc(fb);
  }
  if (lane == 0) waveTot[wv] = cnt;
  __syncthreads();
  if (tid == 0) {
    unsigned run = 0;
    for (int w = 0; w < 8; ++w) { waveExcl[w] = run; run += waveTot[w]; }
  }
  __syncthreads();

  unsigned blockBase = blockBaseScan[blockIdx.x];
#pragma unroll
  for (int v = 0; v < RS_VT; ++v) {
    if ((okbits >> v) & 1u) {
      long long j = base + (long long)wv * (32 * RS_VT) + (long long)v * 32 + lane;
      long long seg = (long long)blockBase + waveExcl[wv] + inc[v] - 1;
      unsigned k = idx[j];
      float ev = edge_value[k];                 // 40 MB, L2-resident gather
      unsigned c = k / (unsigned)E_EDGES;
      atomicAdd(&S[(long long)c * K_TOT + seg], ev);
      if ((flagbits >> v) & 1u) {
        unsigned long long okey = keyv[v];
        unsigned long long row = okey / (unsigned long long)N_NODES;
        unsigned col = (unsigned)(okey - row * (unsigned long long)N_NODES);
        outIndex[seg]         = (int)row;
        outIndex[K_TOT + seg] = (int)col;
      }
    }
  }
}

// ---------------------------------------------------------------------------
// 5) vals = softmax(W)(4x5, padded 16x8) @ S(5xK, padded 8xK) via
//    v_wmma_f32_16x16x4_f32, one 16-column tile per wave, two k-steps.
// ---------------------------------------------------------------------------
__global__ void gt_wmma_gemm_kernel(const float* __restrict__ S,
                                    const float* __restrict__ wmat,
                                    float* __restrict__ vals) {
  int wv = threadIdx.x >> 5, lane = threadIdx.x & 31;
  long long tile = (long long)blockIdx.x * 8 + wv;
  if (tile >= K_TOT / 16) return;            // wave-uniform guard (EXEC all-1 for WMMA)
  long long col0 = tile * 16;

  int m  = lane & 15;        // A: row index
  int kh = lane >> 4;        // k-half: lanes 0-15 hold K=0,1 ; lanes 16-31 hold K=2,3
  int ks = kh * 2;

  // A = softmax(W) padded: rows 4..15 zero; k-step0 covers c=0..3, step1 c=4 (5..7 zero)
  v2f a0, a1;
  a0.x = (m < 4) ? wmat[m * C_IN + ks]     : 0.f;
  a0.y = (m < 4) ? wmat[m * C_IN + ks + 1] : 0.f;
  a1.x = (m < 4 && kh == 0) ? wmat[m * C_IN + 4] : 0.f;
  a1.y = 0.f;

  // B = S tile: lane holds column (lane&15), rows (2*kh, 2*kh+1) per VGPR
  const float* col = S + col0 + (lane & 15);
  v2f b0, b1;
  b0.x = col[(long long)(2 * kh + 0) * K_TOT];
  b0.y = col[(long long)(2 * kh + 1) * K_TOT];
  b1.x = (kh == 0) ? col[(long long)4 * K_TOT] : 0.f;   // row 4 real; rows 5..7 zero
  b1.y = 0.f;

  v8f acc = {};
  acc = __builtin_amdgcn_wmma_f32_16x16x4_f32(false, a0, false, b0, (short)0, acc, false, false);
  acc = __builtin_amdgcn_wmma_f32_16x16x4_f32(false, a1, false, b1, (short)0, acc, false, false);

  // D layout: VGPR r -> M=r (lanes 0-15, N=lane); rows 0..3 are the real outputs
  if (lane < 16) {
    float* o = vals + col0 + lane;
    o[0]         = acc[0];
    o[K_TOT]     = acc[1];
    o[2 * K_TOT] = acc[2];
    o[3 * K_TOT] = acc[3];
  }
}

// ---------------------------------------------------------------------------
extern "C" void kernel_launch(void* const* d_in, const int* in_sizes, int n_in,
                              void* d_out, int out_size, void* d_ws, size_t ws_size,
                              hipStream_t stream) {
  (void)in_sizes; (void)n_in; (void)out_size; (void)ws_size;

  const int*   edge_index = (const int*)d_in[0];
  const float* edge_value = (const float*)d_in[1];
  const float* weight     = (const float*)d_in[2];
  // d_in[3] = num_nodes (100000), baked in as compile-time constant.

  char* ws = (char*)d_ws;
  unsigned long long* keys0 = (unsigned long long*)(ws + OFF_KEYS0);
  unsigned*           idx0  = (unsigned*)(ws + OFF_IDX0);
  unsigned long long* keys1 = (unsigned long long*)(ws + OFF_KEYS1);
  unsigned*           idx1  = (unsigned*)(ws + OFF_IDX1);
  float*              Smat  = (float*)(ws + OFF_S);       // aliases keys1/idx1 (dead after sort)
  unsigned*           hist  = (unsigned*)(ws + OFF_HIST);
  unsigned*           aux   = (unsigned*)(ws + OFF_AUX);
  unsigned*           segc  = (unsigned*)(ws + OFF_SEGCNT);
  unsigned*           segaux= (unsigned*)(ws + OFF_SEGAUX);
  float*              w_ws  = (float*)(ws + OFF_WWS);

  int*   outIndex = (int*)d_out;                          // 2*K int32
  float* outVals  = (float*)d_out + 2 * K_TOT;            // 4*K f32
  float* outW     = (float*)d_out + 6 * K_TOT;            // 20  f32

  const int histScanBlocks = (int)((HIST_N + SC_CHUNK - 1) / SC_CHUNK);     // 1221
  const int segScanBlocks  = (RS_BLOCKS + SC_CHUNK - 1) / SC_CHUNK;         // 3

  // 1) softmax + 2) keys
  hipLaunchKernelGGL(gt_softmax_kernel, dim3(1), dim3(64), 0, stream, weight, w_ws, outW);
  hipLaunchKernelGGL(gt_keygen_kernel, dim3((unsigned)((K_TOT + 255) / 256)), dim3(256), 0, stream,
                     edge_index, keys0, idx0);

  // 3) 4-pass stable LSD radix sort (9-bit digits), ping-pong A<->B, ends in A
  unsigned long long* sk = keys0; unsigned* si = idx0;
  unsigned long long* dk = keys1; unsigned* di = idx1;
  for (int p = 0; p < N_PASSES; ++p) {
    int shift = p * RADIX_BITS;
    hipLaunchKernelGGL(gt_radix_hist_kernel, dim3(RS_BLOCKS), dim3(RS_THREADS), 0, stream,
                       sk, hist, shift);
    hipLaunchKernelGGL(gt_scan_block_kernel, dim3(histScanBlocks), dim3(SC_THREADS), 0, stream,
                       hist, hist, aux, HIST_N);
    hipLaunchKernelGGL(gt_scan_aux_kernel, dim3(1), dim3(1024), 0, stream, aux, histScanBlocks);
    hipLaunchKernelGGL(gt_scan_add_kernel, dim3(histScanBlocks), dim3(SC_THREADS), 0, stream,
                       hist, aux, HIST_N);
    hipLaunchKernelGGL(gt_radix_scatter_kernel, dim3(RS_BLOCKS), dim3(RS_THREADS), 0, stream,
                       sk, si, dk, di, hist, shift);
    { unsigned long long* tk = sk; sk = dk; dk = tk; unsigned* ti = si; si = di; di = ti; }
  }
  // after even #passes, sorted data is in keys0/idx0

  // 4) zero S (aliased over dead B buffers) and the index output region
  hipMemsetAsync(Smat, 0, (size_t)C_IN * K_TOT * 4, stream);
  hipMemsetAsync(outIndex, 0, (size_t)2 * K_TOT * 4, stream);

  hipLaunchKernelGGL(gt_segcount_kernel, dim3(RS_BLOCKS), dim3(RS_THREADS), 0, stream,
                     keys0, segc);
  hipLaunchKernelGGL(gt_scan_block_kernel, dim3(segScanBlocks), dim3(SC_THREADS), 0, stream,
                     segc, segc, segaux, (long long)RS_BLOCKS);
  hipLaunchKernelGGL(gt_scan_aux_kernel, dim3(1), dim3(1024), 0, stream, segaux, segScanBlocks);
  hipLaunchKernelGGL(gt_scan_add_kernel, dim3(segScanBlocks), dim3(SC_THREADS), 0, stream,
                     segc, segaux, (long long)RS_BLOCKS);
  hipLaunchKernelGGL(gt_final_scatter_kernel, dim3(RS_BLOCKS), dim3(RS_THREADS), 0, stream,
                     keys0, idx0, edge_value, segc, Smat, outIndex);

  // 5) vals = softmax(W) @ S via WMMA, 16 columns per wave
  hipLaunchKernelGGL(gt_wmma_gemm_kernel, dim3((unsigned)(K_TOT / 16 / 8)), dim3(256), 0, stream,
                     Smat, w_ws, outVals);
}